// SSKernelNPLR_66374424593125
// MI455X (gfx1250) — compile-verified
//
#include <hip/hip_runtime.h>
#include <hip/hip_bf16.h>
#include <math.h>

typedef __attribute__((ext_vector_type(2))) float v2f;
typedef __attribute__((ext_vector_type(8))) float v8f;

#define HDIM 256
#define NDIM 64
#define LDIM 2048
#define FDIM 1025          // L/2 + 1
#define KUSE 2052          // 2*FDIM padded to multiple of 4 (compute columns)
#define LDK  2056          // A row stride: +4 pad columns so prefetch stays in-bounds
#define TSTEPS (KUSE / 4)  // 513 WMMA K-steps

__device__ inline float2 cmul(float2 a, float2 b) {
    return make_float2(a.x * b.x - a.y * b.y, a.x * b.y + a.y * b.x);
}
__device__ inline float2 cdiv(float2 a, float2 b) {
    float inv = 1.0f / (b.x * b.x + b.y * b.y);
    return make_float2((a.x * b.x + a.y * b.y) * inv, (a.y * b.x - a.x * b.y) * inv);
}

// ---------------------------------------------------------------------------
// Stage 1: Cauchy kernel + rank-1 Woodbury correction.
// One block per h (256 blocks x 256 threads). Threads 0..63 stage per-n
// numerators and w*dt in LDS; each thread then sweeps f = tid, tid+256, ...
// Writes the GEMM A-matrix row h: col[2f] = s_f*Re(kf)/L, col[2f+1] = -s_f*Im(kf)/L.
// ---------------------------------------------------------------------------
__global__ __launch_bounds__(256) void cauchy_woodbury_kernel(
    const float* __restrict__ C, const float* __restrict__ B,
    const float* __restrict__ P, const float* __restrict__ w,
    const float* __restrict__ log_dt, float* __restrict__ A) {
    __shared__ float2 s_v00[NDIM], s_v01[NDIM], s_v10[NDIM], s_v11[NDIM];
    __shared__ float2 s_wdt[NDIM];

    const int h = blockIdx.x;
    const int tid = threadIdx.x;
    const float dt = __expf(log_dt[h]);

    if (tid < NDIM) {
        const int n = tid;
        const int base = (h * NDIM + n) * 2;
        float2 Cc = make_float2(C[base], C[base + 1]);
        float2 Bc = make_float2(B[base], B[base + 1]);
        float2 Pc = make_float2(P[base], P[base + 1]);
        float2 wc = make_float2(w[base], w[base + 1]);
        float2 Qc = make_float2(Pc.x, -Pc.y);   // conj(P)
        s_v00[n] = cmul(Bc, Cc);
        s_v01[n] = cmul(Bc, Qc);
        s_v10[n] = cmul(Pc, Cc);
        s_v11[n] = cmul(Pc, Qc);
        s_wdt[n] = make_float2(wc.x * dt, wc.y * dt);
    }
    __syncthreads();

    const float two_pi_over_L = 6.28318530717958647692f / (float)LDIM;

    for (int f = tid; f < FDIM; f += 256) {
        // omega = exp(-2*pi*i*f/L), z = 2(1-omega)/(1+omega)
        float sw, cw;
        __sincosf(two_pi_over_L * (float)f, &sw, &cw);
        float2 om = make_float2(cw, -sw);
        float2 one_m = make_float2(1.0f - om.x, -om.y);
        float2 one_p = make_float2(1.0f + om.x, om.y);
        float2 z = cdiv(make_float2(2.0f * one_m.x, 2.0f * one_m.y), one_p);

        float2 r00 = make_float2(0.f, 0.f), r01 = make_float2(0.f, 0.f);
        float2 r10 = make_float2(0.f, 0.f), r11 = make_float2(0.f, 0.f);
#pragma unroll 8
        for (int n = 0; n < NDIM; ++n) {
            float2 wd = s_wdt[n];
            float2 d = make_float2(z.x - wd.x, z.y - wd.y);
            float inv = 1.0f / (d.x * d.x + d.y * d.y);
            float2 di = make_float2(d.x * inv, -d.y * inv);  // 1/(z - w*dt)
            float2 t;
            t = cmul(s_v00[n], di); r00.x += t.x; r00.y += t.y;
            t = cmul(s_v01[n], di); r01.x += t.x; r01.y += t.y;
            t = cmul(s_v10[n], di); r10.x += t.x; r10.y += t.y;
            t = cmul(s_v11[n], di); r11.x += t.x; r11.y += t.y;
        }
        r00.x *= dt; r00.y *= dt; r01.x *= dt; r01.y *= dt;
        r10.x *= dt; r10.y *= dt; r11.x *= dt; r11.y *= dt;

        // kf = r00 - r01*r10/(1+r11), then * 2/(1+omega)
        float2 denom = make_float2(1.0f + r11.x, r11.y);
        float2 corr = cdiv(cmul(r01, r10), denom);
        float2 kf = make_float2(r00.x - corr.x, r00.y - corr.y);
        kf = cmul(kf, cdiv(make_float2(2.0f, 0.0f), one_p));

        // irfft folding weight: 1 for DC/Nyquist, 2 otherwise; 1/L overall
        float sc = ((f == 0) || (f == LDIM / 2)) ? 1.0f : 2.0f;
        sc *= (1.0f / (float)LDIM);
        A[(size_t)h * LDK + 2 * f]     = sc * kf.x;
        A[(size_t)h * LDK + 2 * f + 1] = -sc * kf.y;
    }
    if (tid == 0) {   // zero K-pad + prefetch-pad columns
        float* row = A + (size_t)h * LDK;
        row[2050] = 0.0f; row[2051] = 0.0f;
        row[2052] = 0.0f; row[2053] = 0.0f; row[2054] = 0.0f; row[2055] = 0.0f;
    }
}

// ---------------------------------------------------------------------------
// Stage 2: irfft as real GEMM via V_WMMA_F32_16X16X4_F32.
//   out(256 x 2048) = A(256 x 2052) * Bdft(2052 x 2048)
//   Bdft[2f][l] = cos(2*pi*f*l/L), Bdft[2f+1][l] = sin(2*pi*f*l/L)
// generated on the fly (exact mod-2048 reduction, one __sincosf/lane/step).
//
// Block = 4 waves owning 4 M-tiles x 1 N-tile; the 513 K-steps are K-split
// across the 4 waves (scalar chunk bounds via readfirstlane) into private
// accumulators, combined via an LDS reduction -> 2048 waves for occupancy.
// 1-deep register double-buffering + unroll-2 keeps up to 8 A-fragment loads
// in flight behind the WMMAs.
//
// Fragment layouts (ISA 7.12.2):
//  A 16x4 f32 (2 VGPR): lanes 0-15 = (M=lane, K=4t,4t+1); lanes 16-31 = K=4t+2,4t+3.
//  B 4x16 f32 (2 VGPR): lanes 0-15 = rows {cos,sin} of f=2t; lanes 16-31 = f=2t+1.
//  C/D 16x16 f32 (8 VGPR): vgpr r -> row r (lanes 0-15) / row r+8 (lanes 16-31).
// ---------------------------------------------------------------------------
__global__ __launch_bounds__(128) void irfft_wmma_gemm(
    const float* __restrict__ A, float* __restrict__ out) {
    __shared__ float red[3][32][32];   // partial sums from waves 1..3

    const int tid = threadIdx.x;
    const int lane = tid & 31;
    // force wave id into an SGPR: K-chunk bounds become provably scalar, so
    // the K-loop compiles to s_cmp/s_cbranch (no EXEC masking around WMMA)
    const int wv = __builtin_amdgcn_readfirstlane(tid >> 5);
    const int ntile = blockIdx.x & 127;       // 128 N-tiles of 16
    const int mgroup = blockIdx.x >> 7;       // 4 groups of 4 M-tiles
    const int m0 = mgroup * 64;
    const int n0 = ntile * 16;
    const int lr = lane & 15;
    const int hi = (lane >> 4) & 1;
    const int ncol = n0 + lr;                 // global column this lane builds

    // K-chunk for this wave: [t0, t1) of the 513 steps (129,129,129,126)
    const int t0 = wv * 129;
    const int t1 = (wv == 3) ? TSTEPS : (t0 + 129);

    const float two_pi_over_L = 6.28318530717958647692f / (float)LDIM;

    v8f acc0 = {}, acc1 = {}, acc2 = {}, acc3 = {};

    const float* a0p = A + (size_t)(m0 + lr) * LDK;
    const float* a1p = A + (size_t)(m0 + 16 + lr) * LDK;
    const float* a2p = A + (size_t)(m0 + 32 + lr) * LDK;
    const float* a3p = A + (size_t)(m0 + 48 + lr) * LDK;

    // pipeline prologue: fragments for step t0
    int col = 4 * t0 + 2 * hi;
    v2f a0 = *(const v2f*)(a0p + col);
    v2f a1 = *(const v2f*)(a1p + col);
    v2f a2 = *(const v2f*)(a2p + col);
    v2f a3 = *(const v2f*)(a3p + col);

#pragma unroll 2
    for (int t = t0; t < t1; ++t) {
        // prefetch next step's fragments (t+1 <= 513; LDK pad keeps it in-bounds)
        const int coln = 4 * (t + 1) + 2 * hi;
        v2f na0 = *(const v2f*)(a0p + coln);
        v2f na1 = *(const v2f*)(a1p + coln);
        v2f na2 = *(const v2f*)(a2p + coln);
        v2f na3 = *(const v2f*)(a3p + coln);

        // B fragment: exact range reduction (f*l mod 2048), one sincos
        const int f = 2 * t + hi;
        const int m = (f * ncol) & (LDIM - 1);
        float s, c;
        __sincosf(two_pi_over_L * (float)m, &s, &c);
        const float valid = (f <= LDIM / 2) ? 1.0f : 0.0f;  // mask K-pad rows
        v2f bfrag; bfrag.x = c * valid; bfrag.y = s * valid;

        acc0 = __builtin_amdgcn_wmma_f32_16x16x4_f32(
            false, a0, false, bfrag, (short)0, acc0, false, false);
        acc1 = __builtin_amdgcn_wmma_f32_16x16x4_f32(
            false, a1, false, bfrag, (short)0, acc1, false, false);
        acc2 = __builtin_amdgcn_wmma_f32_16x16x4_f32(
            false, a2, false, bfrag, (short)0, acc2, false, false);
        acc3 = __builtin_amdgcn_wmma_f32_16x16x4_f32(
            false, a3, false, bfrag, (short)0, acc3, false, false);

        a0 = na0; a1 = na1; a2 = na2; a3 = na3;
    }

    // K-split reduction across the 4 waves via LDS
    if (wv > 0) {
#pragma unroll
        for (int e = 0; e < 8; ++e) {
            red[wv - 1][lane][e]      = acc0[e];
            red[wv - 1][lane][8 + e]  = acc1[e];
            red[wv - 1][lane][16 + e] = acc2[e];
            red[wv - 1][lane][24 + e] = acc3[e];
        }
    }
    __syncthreads();
    if (wv == 0) {
#pragma unroll
        for (int wp = 0; wp < 3; ++wp) {
#pragma unroll
            for (int e = 0; e < 8; ++e) {
                acc0[e] += red[wp][lane][e];
                acc1[e] += red[wp][lane][8 + e];
                acc2[e] += red[wp][lane][16 + e];
                acc3[e] += red[wp][lane][24 + e];
            }
        }
        const int row_off = hi ? 8 : 0;
#pragma unroll
        for (int r = 0; r < 8; ++r) {
            const int grow = m0 + r + row_off;
            out[(size_t)(grow)      * LDIM + n0 + lr] = acc0[r];
            out[(size_t)(grow + 16) * LDIM + n0 + lr] = acc1[r];
            out[(size_t)(grow + 32) * LDIM + n0 + lr] = acc2[r];
            out[(size_t)(grow + 48) * LDIM + n0 + lr] = acc3[r];
        }
    }
}

// ---------------------------------------------------------------------------
// Host launcher. Inputs (setup_inputs order): C, B, P, w, log_dt, L.
// d_ws holds the A matrix: 256 * 2056 * 4 bytes ~= 2.1 MB.
// ---------------------------------------------------------------------------
extern "C" void kernel_launch(void* const* d_in, const int* in_sizes, int n_in,
                              void* d_out, int out_size, void* d_ws, size_t ws_size,
                              hipStream_t stream) {
    const float* C = (const float*)d_in[0];
    const float* B = (const float*)d_in[1];
    const float* P = (const float*)d_in[2];
    const float* w = (const float*)d_in[3];
    const float* log_dt = (const float*)d_in[4];
    float* Amat = (float*)d_ws;
    float* out = (float*)d_out;

    cauchy_woodbury_kernel<<<HDIM, 256, 0, stream>>>(C, B, P, w, log_dt, Amat);

    // 4 M-groups x 128 N-tiles, 4 K-split waves per block -> 2048 waves
    irfft_wmma_gemm<<<4 * 128, 128, 0, stream>>>(Amat, out);
}